// DeformConv2d_17867063952056
// MI455X (gfx1250) — compile-verified
//
#include <hip/hip_runtime.h>
#include <math.h>

// Problem constants (from reference): B=8, CIN=64, COUT=64, H=W=128, K=3, PAD=1
#define Bn   8
#define CIN  64
#define COUT 64
#define Hd   128
#define Wd   128
#define KKn  9
#define KTOT (CIN*KKn)     // 576 = contraction length
#define TP   16            // pixels per tile
#define ASTR (KTOT+2)      // padded LDS stride (578) to spread banks

typedef __attribute__((ext_vector_type(2))) float v2f;
typedef __attribute__((ext_vector_type(8))) float v8f;

__device__ __forceinline__ v8f wmma_f32_k4(v2f a, v2f b, v8f c) {
  // V_WMMA_F32_16X16X4_F32 : D(16x16 f32) = A(16x4 f32) * B(4x16 f32) + C
  // 8-arg pattern: (neg_a, A, neg_b, B, c_mod, C, reuse_a, reuse_b)
  return __builtin_amdgcn_wmma_f32_16x16x4_f32(false, a, false, b, (short)0, c,
                                               false, false);
}

__global__ __launch_bounds__(128) void dcn_fused(
    const float* __restrict__ x, const float* __restrict__ w_off,
    const float* __restrict__ b_off, const float* __restrict__ w_dcn,
    float* __restrict__ out)
{
  __shared__ __align__(16) float a_lds[TP][ASTR];     // im2col tile: 36992 B
  __shared__ float dPY[KKn][TP];                      // per-tap sample y
  __shared__ float dPX[KKn][TP];                      // per-tap sample x
  __shared__ float dM [KKn][TP];                      // per-tap mask

  const int tile = blockIdx.x;            // 8192 tiles
  const int w0   = (tile & 7) << 4;       // 16-pixel strip along W
  const int h    = (tile >> 3) & (Hd - 1);
  const int b    = tile >> 10;
  const int tid  = threadIdx.x;
  const int wave = tid >> 5;
  const int lane = tid & 31;
  const int nIdx = lane & 15;             // A: pixel row / B: cout column
  const int koff = (lane >> 4) << 1;      // lanes 16-31 hold K+2,K+3
  const int pbase = (lane >> 4) << 3;     // C layout: reg r -> pixel r (+8)

  const float* xb = x + (size_t)b * CIN * Hd * Wd;

  // ---------- P1: plain im2col tile for the offset conv ----------
  for (int e = tid; e < TP * KTOT; e += 128) {
    int p = e & 15, kidx = e >> 4;        // kidx = c*9 + kk (w_off K order)
    int c = kidx / KKn, kk = kidx - c * KKn;
    int ky = kk / 3, kx = kk - ky * 3;
    int yy = h + ky - 1;
    int xx = w0 + p + kx - 1;
    float v = 0.f;
    if (yy >= 0 && yy < Hd && xx >= 0 && xx < Wd)
      v = xb[c * Hd * Wd + yy * Wd + xx];
    a_lds[p][kidx] = v;
  }
  __syncthreads();

  // ---------- P2: offset conv GEMM (waves 0-1, 27 couts padded to 32) ----------
  if (wave < 2) {
    int coutT = (wave << 4) + nIdx;                 // 0..31 (27..31 dead lanes)
    int coutC = coutT < 27 ? coutT : 26;            // clamp B row, never stored
    const float* Brow = w_off + (size_t)coutC * KTOT;
    float bias = b_off[coutC];
    v8f acc = {bias, bias, bias, bias, bias, bias, bias, bias};
    #pragma unroll 4
    for (int kc = 0; kc < KTOT; kc += 4) {
      int kb = kc + koff;
      v2f a  = *(const v2f*)&a_lds[nIdx][kb];       // ds_load_b64
      v2f bb = *(const v2f*)&Brow[kb];              // L2-resident weights
      acc = wmma_f32_k4(a, bb, acc);
    }
    if (coutT < 27) {
      // off[2k]=dy, off[2k+1]=dx, off[18+k]=mask logit (reference mapping)
      #pragma unroll
      for (int r = 0; r < 8; ++r) {
        int p = pbase + r;                          // pixel owned by this reg
        float v = acc[r];
        if (coutT >= 18) {
          dM[coutT - 18][p] = 1.f / (1.f + __expf(-v));
        } else if (!(coutT & 1)) {
          int kk = coutT >> 1;
          dPY[kk][p] = (float)(h + kk / 3 - 1) + v;           // h + ki + dy
        } else {
          int kk = coutT >> 1;
          dPX[kk][p] = (float)(w0 + p + (kk - (kk / 3) * 3) - 1) + v; // w + kj + dx
        }
      }
    }
  }
  __syncthreads();

  // ---------- P3: bilinear-gathered, mask-scaled im2col (the hot phase) ----------
  for (int e = tid; e < TP * KTOT; e += 128) {
    int p = e & 15, kidx = e >> 4;
    int c = kidx / KKn, kk = kidx - c * KKn;
    float py = dPY[kk][p], px = dPX[kk][p], m = dM[kk][p];
    float fy = floorf(py), fx = floorf(px);
    int y0 = (int)fy, x0 = (int)fx;
    float wy1 = py - fy, wx1 = px - fx;
    float wy0 = 1.f - wy1, wx0 = 1.f - wx1;
    const float* xc = xb + c * Hd * Wd;
    bool y0o = (y0 >= 0) & (y0 < Hd);
    bool y1o = (y0 + 1 >= 0) & (y0 + 1 < Hd);
    bool x0o = (x0 >= 0) & (x0 < Wd);
    bool x1o = (x0 + 1 >= 0) & (x0 + 1 < Wd);
    float v = 0.f;
    if (y0o & x0o) v += wy0 * wx0 * xc[y0 * Wd + x0];
    if (y0o & x1o) v += wy0 * wx1 * xc[y0 * Wd + x0 + 1];
    if (y1o & x0o) v += wy1 * wx0 * xc[(y0 + 1) * Wd + x0];
    if (y1o & x1o) v += wy1 * wx1 * xc[(y0 + 1) * Wd + x0 + 1];
    a_lds[p][kidx] = v * m;
  }
  __syncthreads();

  // ---------- P4: main deformable GEMM (4 waves x 16 couts) ----------
  {
    int coutT = (wave << 4) + nIdx;                 // 0..63
    const float* Brow = w_dcn + (size_t)coutT * KTOT; // [o][c*9+kk] contiguous
    v8f acc = {};
    #pragma unroll 4
    for (int kc = 0; kc < KTOT; kc += 4) {
      int kb = kc + koff;
      v2f a  = *(const v2f*)&a_lds[nIdx][kb];
      v2f bb = *(const v2f*)&Brow[kb];
      acc = wmma_f32_k4(a, bb, acc);
    }
    // C layout: reg r, lanes 0-15 -> pixel r; lanes 16-31 -> pixel r+8.
    // Consecutive regs = consecutive pixels => two float4 stores per lane.
    float* orow = out + (((size_t)b * COUT + coutT) * Hd + h) * Wd + w0 + pbase;
    *(float4*)&orow[0] = make_float4(acc[0], acc[1], acc[2], acc[3]);
    *(float4*)&orow[4] = make_float4(acc[4], acc[5], acc[6], acc[7]);
  }
}

extern "C" void kernel_launch(void* const* d_in, const int* in_sizes, int n_in,
                              void* d_out, int out_size, void* d_ws, size_t ws_size,
                              hipStream_t stream) {
  (void)in_sizes; (void)n_in; (void)d_ws; (void)ws_size; (void)out_size;
  const float* x     = (const float*)d_in[0];
  const float* w_off = (const float*)d_in[1];
  const float* b_off = (const float*)d_in[2];
  const float* w_dcn = (const float*)d_in[3];
  float* out = (float*)d_out;
  dim3 grid(Bn * Hd * (Wd / TP));   // 8 * 128 * 8 = 8192 tiles
  dcn_fused<<<grid, 128, 0, stream>>>(x, w_off, b_off, w_dcn, out);
}